// KnowledgeCriterion_82102594831164
// MI455X (gfx1250) — compile-verified
//
#include <hip/hip_runtime.h>
#include <hip/hip_bf16.h>
#include <math.h>

// Problem constants (from setup_inputs): B=256, S=64, R=128, F=1536, D=256
#define B_ 256
#define S_ 64
#define R_ 128
#define D_ 256
#define F_ 1536
#define BR_ (B_ * R_)          // 32768
#define BSR_ (B_ * S_ * R_)    // 2097152
#define DENOM_ 536870912.0f    // B*S*R*D

typedef __attribute__((ext_vector_type(2))) float v2f;
typedef __attribute__((ext_vector_type(8))) float v8f;

#if defined(__has_builtin)
#if __has_builtin(__builtin_amdgcn_wmma_f32_16x16x4_f32)
#define HAVE_WMMA4 1
#endif
#endif

__device__ __forceinline__ float wave_sum32(float v) {
#pragma unroll
    for (int off = 16; off > 0; off >>= 1) v += __shfl_xor(v, off, 32);
    return v;
}

// ---------------------------------------------------------------------------
// Kernel A: one wave per (b,r) row of tri_feat (1536 f32 = 6 slices of 256).
// Produces s0[b*R+r] and sumsq[b*R+r].  201 MB streamed -> HBM-bound.
// ---------------------------------------------------------------------------
__global__ void k_s0_sumsq(const float* __restrict__ tri,
                           float* __restrict__ s0_out,
                           float* __restrict__ ss_out) {
    const int wave = threadIdx.x >> 5;
    const int lane = threadIdx.x & 31;
    const int idx  = blockIdx.x * 8 + wave;      // 0 .. B*R-1
    const float* base = tri + (size_t)idx * F_;

    // Prefetch the row this wave will NOT reuse but the cacheline stream ahead:
    // speculative prefetch of the row 8 rows ahead (next block, same wave slot).
    __builtin_prefetch(base + 8 * F_ + lane * 16, 0, 0);

    float s0 = 0.f, ss = 0.f;
#pragma unroll
    for (int c = 0; c < 2; ++c) {
        const int o = c * 128 + lane * 4;        // element offset inside a slice
        float4 hre = *(const float4*)(base + 0 * D_ + o);
        float4 him = *(const float4*)(base + 1 * D_ + o);
        float4 rre = *(const float4*)(base + 2 * D_ + o);
        float4 rim = *(const float4*)(base + 3 * D_ + o);
        float4 tre = *(const float4*)(base + 4 * D_ + o);
        float4 tim = *(const float4*)(base + 5 * D_ + o);
#define ACC(q_)                                                              \
        {                                                                    \
            float p = hre.q_ * tre.q_ + him.q_ * tim.q_;                     \
            float q = hre.q_ * tim.q_ - him.q_ * tre.q_;                     \
            s0 += rre.q_ * p + rim.q_ * q;                                   \
            ss += hre.q_ * hre.q_ + him.q_ * him.q_ + rre.q_ * rre.q_ +      \
                  rim.q_ * rim.q_ + tre.q_ * tre.q_ + tim.q_ * tim.q_;       \
        }
        ACC(x) ACC(y) ACC(z) ACC(w)
#undef ACC
    }
    s0 = wave_sum32(s0);
    ss = wave_sum32(ss);
    if (lane == 0) { s0_out[idx] = s0; ss_out[idx] = ss; }
}

// ---------------------------------------------------------------------------
// Kernel B: a2s[b,r] = sum_s ((alpha[b,s,r]-0.1)*mask[b,s])^2
// One wave per (b, 16-column tile).  Column sum of a 64x16 slab expressed as
// ones(16x4) @ B(4x16) accumulated 16 times via V_WMMA_F32_16X16X4_F32.
// With A == all-ones the result is invariant to the K-slot permutation of the
// B-operand VGPR layout (each of the 4 rows is counted exactly once).
// ---------------------------------------------------------------------------
__global__ void k_a2s(const float* __restrict__ alpha,
                      const float* __restrict__ mask,
                      float* __restrict__ a2s) {
    const int wave = threadIdx.x >> 5;           // column tile 0..7
    const int lane = threadIdx.x & 31;
    const int b    = blockIdx.x;                 // 0..255
    const int half = lane >> 4;                  // 0 or 1
    const int n    = lane & 15;
    const int col  = wave * 16 + n;              // 0..127

#if defined(HAVE_WMMA4)
    v8f c = {};
    v2f ones; ones[0] = 1.0f; ones[1] = 1.0f;
#pragma unroll 4
    for (int t = 0; t < 16; ++t) {
        const int sb  = t * 4;
        const int sA  = sb + half;               // rows sb..sb+3 covered by the
        const int sB  = sb + 2 + half;           // two halves x two B VGPRs
        float m0 = mask[b * S_ + sA];
        float m1 = mask[b * S_ + sB];
        float x0 = (alpha[(size_t)(b * S_ + sA) * R_ + col] - 0.1f) * m0;
        float x1 = (alpha[(size_t)(b * S_ + sB) * R_ + col] - 0.1f) * m1;
        v2f bv; bv[0] = x0 * x0; bv[1] = x1 * x1;
        c = __builtin_amdgcn_wmma_f32_16x16x4_f32(
                /*neg_a=*/false, ones, /*neg_b=*/false, bv,
                /*c_mod=*/(short)0, c, /*reuse_a=*/false, /*reuse_b=*/false);
    }
    if (lane < 16) a2s[b * R_ + col] = c[0];     // every D row holds the colsum
#else
    if (lane < 16) {
        float acc = 0.f;
        for (int s = 0; s < S_; ++s) {
            float m = mask[b * S_ + s];
            float x = (alpha[(size_t)(b * S_ + s) * R_ + col] - 0.1f) * m;
            acc += x * x;
        }
        a2s[b * R_ + col] = acc;
    }
#endif
}

// ---------------------------------------------------------------------------
// Kernel N: numtrue = sum(mask); also zero the softplus accumulator (ws[1]).
// ---------------------------------------------------------------------------
__global__ void k_numtrue(const float* __restrict__ mask, float* __restrict__ ws) {
    float v = 0.f;
    for (int i = threadIdx.x; i < B_ * S_; i += 256) v += mask[i];
    v = wave_sum32(v);
    __shared__ float part[8];
    if ((threadIdx.x & 31) == 0) part[threadIdx.x >> 5] = v;
    __syncthreads();
    if (threadIdx.x == 0) {
        float tot = 0.f;
#pragma unroll
        for (int i = 0; i < 8; ++i) tot += part[i];
        ws[0] = tot;
        ws[1] = 0.f;   // reset softplus accumulator for this call
    }
}

// ---------------------------------------------------------------------------
// Kernel C: sum over (b,s,r) of softplus(-a^3 * s0[b,r]); atomicAdd into ws[1].
// ---------------------------------------------------------------------------
__global__ void k_score(const float* __restrict__ alpha,
                        const float* __restrict__ mask,
                        const float* __restrict__ s0,
                        float* __restrict__ ws) {
    const int t = blockIdx.x * 256 + threadIdx.x;      // exact: 8192*256 = BSR_
    const int r = t & (R_ - 1);
    const int s = (t >> 7) & (S_ - 1);
    const int b = t >> 13;
    float m = mask[b * S_ + s];
    float a = (alpha[t] - 0.1f) * m;
    float x = -(a * a * a) * s0[b * R_ + r];
    // stable softplus: max(x,0) + log1p(exp(-|x|))
    float sp = fmaxf(x, 0.f) + log1pf(__expf(-fabsf(x)));

    sp = wave_sum32(sp);
    __shared__ float part[8];
    if ((threadIdx.x & 31) == 0) part[threadIdx.x >> 5] = sp;
    __syncthreads();
    if (threadIdx.x == 0) {
        float tot = 0.f;
#pragma unroll
        for (int i = 0; i < 8; ++i) tot += part[i];
        atomicAdd(&ws[1], tot);
    }
}

// ---------------------------------------------------------------------------
// Kernel D: regul dot = sum_{b,r} a2s * sumsq  (32768 elems, single block).
// ---------------------------------------------------------------------------
__global__ void k_dot(const float* __restrict__ a2s,
                      const float* __restrict__ ss,
                      float* __restrict__ ws) {
    float v = 0.f;
    for (int i = threadIdx.x; i < BR_; i += 256) v += a2s[i] * ss[i];
    v = wave_sum32(v);
    __shared__ float part[8];
    if ((threadIdx.x & 31) == 0) part[threadIdx.x >> 5] = v;
    __syncthreads();
    if (threadIdx.x == 0) {
        float tot = 0.f;
#pragma unroll
        for (int i = 0; i < 8; ++i) tot += part[i];
        ws[2] = tot;
    }
}

// ---------------------------------------------------------------------------
// Kernel E: out = score_sum/numtrue + 0.01 * (dot / (B*S*R*D))
// ---------------------------------------------------------------------------
__global__ void k_final(const float* __restrict__ ws, float* __restrict__ out) {
    out[0] = ws[1] / ws[0] + 0.01f * (ws[2] / DENOM_);
}

extern "C" void kernel_launch(void* const* d_in, const int* in_sizes, int n_in,
                              void* d_out, int out_size, void* d_ws, size_t ws_size,
                              hipStream_t stream) {
    const float* tri   = (const float*)d_in[0];   // (B,R,F) f32
    const float* alpha = (const float*)d_in[1];   // (B,S,R) f32
    const float* mask  = (const float*)d_in[2];   // (B,S)   f32
    float* out = (float*)d_out;

    float* ws   = (float*)d_ws;          // [0]=numtrue [1]=score_sum [2]=dot
    float* s0   = ws + 16;               // B*R
    float* ss   = s0 + BR_;              // B*R
    float* a2s  = ss + BR_;              // B*R

    (void)in_sizes; (void)n_in; (void)out_size; (void)ws_size;

    k_s0_sumsq<<<BR_ / 8, 256, 0, stream>>>(tri, s0, ss);
    k_a2s     <<<B_,      256, 0, stream>>>(alpha, mask, a2s);
    k_numtrue <<<1,       256, 0, stream>>>(mask, ws);
    k_score   <<<BSR_ / 256, 256, 0, stream>>>(alpha, mask, s0, ws);
    k_dot     <<<1,       256, 0, stream>>>(a2s, ss, ws);
    k_final   <<<1,       1,   0, stream>>>(ws, out);
}